// GATPool_78245714198781
// MI455X (gfx1250) — compile-verified
//
#include <hip/hip_runtime.h>
#include <math.h>

// ---------------- problem constants (from reference) ----------------
#define NN      20480      // nodes
#define EE      327680     // edges
#define GG      64         // graphs
#define IN_DIM  128
#define HEADS   4
#define DHEAD   64
#define HD      256        // HEADS*DHEAD
#define CLS     10
#define NEG_SLOPE 0.2f
#define NODES_PER_G 320
#define KC      16         // K-chunk staged in LDS per double-buffer slot

typedef __attribute__((ext_vector_type(2))) float v2f;
typedef __attribute__((ext_vector_type(8))) float v8f;

__device__ __forceinline__ float elu1(float x) {
    return x > 0.0f ? x : (__expf(x) - 1.0f);
}

// float atomic max via int-max / uint-min bit trick (works with -inf init)
__device__ __forceinline__ void atomicMaxF32(float* addr, float val) {
    if (val >= 0.0f) {
        atomicMax((int*)addr, __float_as_int(val));
    } else {
        atomicMin((unsigned int*)addr, __float_as_uint(val));
    }
}

// async global->LDS copy of KC*HD floats (16KB) by 128 threads, b128 granules.
// Tracked on ASYNCcnt; caller must s_wait_asynccnt + barrier before reading.
__device__ __forceinline__ void async_copy_chunk(const float* __restrict__ gsrc,
                                                 float* lds_dst) {
    const int t = threadIdx.x;                      // 0..127
    unsigned lbase = (unsigned)(size_t)lds_dst;     // LDS byte offset of buffer
#pragma unroll
    for (int i = 0; i < (KC * HD) / (128 * 4); ++i) {   // 8 x b128 per thread
        int off = t * 4 + i * 512;                      // float offset
        unsigned laddr = lbase + (unsigned)(off * 4);
        const float* g = gsrc + off;
        asm volatile("global_load_async_to_lds_b128 %0, %1, off"
                     :: "v"(laddr), "v"(g)
                     : "memory");
    }
}

__device__ __forceinline__ void wait_async_all() {
    asm volatile("s_wait_asynccnt 0x0" ::: "memory");
}

// ---------------- kernels ----------------

__global__ void fill_f32(float* __restrict__ p, float v, int n) {
    int i = blockIdx.x * blockDim.x + threadIdx.x;
    if (i < n) p[i] = v;
}

// C[M,HD] = A[M,K] @ B[K,HD], fp32 WMMA 16x16x4.
// Block = 128 threads = 4 waves, owns one 16-row strip (tm = blockIdx.x).
// Wave w computes columns [w*64, w*64+64) as four 16x16 accumulators so each
// A fragment feeds 4 WMMAs. B panel is streamed through LDS in KC-row chunks
// with async loads double-buffered against the WMMA loop.
__global__ void __launch_bounds__(128)
gemm_wmma_f32(const float* __restrict__ A, const float* __restrict__ B,
              float* __restrict__ C, int M, int K) {
    __shared__ __align__(16) float bsh[2][KC * HD];   // 2 x 16KB

    const int lane = threadIdx.x & 31;
    const int wave = threadIdx.x >> 5;     // 0..3 -> 64-col group
    const int tm   = blockIdx.x;           // 16-row strip
    const int half = lane >> 4;            // 0: K+0/1, 1: K+2/3
    const int lr   = lane & 15;

    const float* arow = A + (size_t)(tm * 16 + lr) * K + 2 * half;

    v8f acc[4] = {v8f{}, v8f{}, v8f{}, v8f{}};

    const int NC = K / KC;
    async_copy_chunk(B, &bsh[0][0]);
    for (int c = 0; c < NC; ++c) {
        const int par = c & 1;
        wait_async_all();                  // my chunk-c b128s have landed
        __syncthreads();                   // everyone's landed; prev reads done
        if (c + 1 < NC)                    // prefetch next chunk into other buf
            async_copy_chunk(B + (size_t)(c + 1) * KC * HD, &bsh[1 - par][0]);

        const float* ab = arow + c * KC;
#pragma unroll
        for (int k = 0; k < KC; k += 4) {
            v2f a = *(const v2f*)(ab + k);                 // A[m, ck+2h : +2]
            const float* brow = &bsh[par][(k + 2 * half) * HD + wave * 64 + lr];
#pragma unroll
            for (int j = 0; j < 4; ++j) {
                v2f b;
                b.x = brow[j * 16];                        // B[ck+2h,   n]
                b.y = brow[j * 16 + HD];                   // B[ck+2h+1, n]
                acc[j] = __builtin_amdgcn_wmma_f32_16x16x4_f32(
                    false, a, false, b, (short)0, acc[j], false, false);
            }
        }
    }

    // C/D layout: VGPR v -> row (v + 8*half), col = lr
    float* crow = C + (size_t)(tm * 16 + 8 * half) * HD + wave * 64 + lr;
#pragma unroll
    for (int j = 0; j < 4; ++j)
#pragma unroll
        for (int v = 0; v < 8; ++v)
            crow[(size_t)v * HD + j * 16] = acc[j][v];
}

// el[n,h] = sum_d z[n,h,d]*al[h,d]; er likewise. One wave32 per (node, head).
__global__ void attn_logits(const float* __restrict__ Z,
                            const float* __restrict__ al,
                            const float* __restrict__ ar,
                            float* __restrict__ el, float* __restrict__ er) {
    int wg   = (blockIdx.x * blockDim.x + threadIdx.x) >> 5;
    int lane = threadIdx.x & 31;
    int node = wg / HEADS;
    int head = wg % HEADS;
    if (node >= NN) return;
    const float* zp  = Z  + (size_t)node * HD + head * DHEAD;
    const float* alp = al + head * DHEAD;
    const float* arp = ar + head * DHEAD;
    float z0 = zp[lane], z1 = zp[lane + 32];
    float sl = z0 * alp[lane] + z1 * alp[lane + 32];
    float sr = z0 * arp[lane] + z1 * arp[lane + 32];
#pragma unroll
    for (int off = 16; off > 0; off >>= 1) {
        sl += __shfl_down(sl, off, 32);
        sr += __shfl_down(sr, off, 32);
    }
    if (lane == 0) {
        el[node * HEADS + head] = sl;
        er[node * HEADS + head] = sr;
    }
}

// e[eid,h] = leaky_relu(el[src]+er[dst]); atomic segment max into m[dst,h]
__global__ void edge_logits_max(const int* __restrict__ src,
                                const int* __restrict__ dst,
                                const float* __restrict__ el,
                                const float* __restrict__ er,
                                float* __restrict__ e, float* __restrict__ m) {
    int idx = blockIdx.x * blockDim.x + threadIdx.x;       // over E*H
    if (idx >= EE * HEADS) return;
    int eid = idx >> 2;
    int h   = idx & 3;
    int s = src[eid], d = dst[eid];
    float v = el[s * HEADS + h] + er[d * HEADS + h];
    v = v > 0.0f ? v : NEG_SLOPE * v;
    e[idx] = v;
    atomicMaxF32(&m[d * HEADS + h], v);
}

// ex = exp(e - m[dst]); e <- ex; atomic segment sum into s[dst,h]
__global__ void edge_exp_sum(const int* __restrict__ dst,
                             const float* __restrict__ m,
                             float* __restrict__ e, float* __restrict__ s) {
    int idx = blockIdx.x * blockDim.x + threadIdx.x;       // over E*H
    if (idx >= EE * HEADS) return;
    int eid = idx >> 2;
    int h   = idx & 3;
    int d = dst[eid];
    float ex = __expf(e[idx] - m[d * HEADS + h]);
    e[idx] = ex;
    atomicAdd(&s[d * HEADS + h], ex);
}

// agg[dst,:] += z[src,:] * a ; one 256-thread block per edge
__global__ void edge_message(const int* __restrict__ src,
                             const int* __restrict__ dst,
                             const float* __restrict__ Z,
                             const float* __restrict__ e,
                             const float* __restrict__ s,
                             float* __restrict__ agg) {
    int eid = blockIdx.x;
    int hd  = threadIdx.x;          // 0..255
    int h   = hd >> 6;
    int sn = src[eid], dn = dst[eid];
    float denom = s[dn * HEADS + h];
    float a = e[eid * HEADS + h] / (denom > 0.0f ? denom : 1.0f);
    atomicAdd(&agg[(size_t)dn * HD + hd], Z[(size_t)sn * HD + hd] * a);
}

// in-place: agg = elu( residual ? elu(agg + hin) : agg )
__global__ void act_residual(const float* __restrict__ hin,
                             float* __restrict__ agg, int residual) {
    size_t i = (size_t)blockIdx.x * blockDim.x + threadIdx.x;  // over N*HD
    if (i >= (size_t)NN * HD) return;
    float v = agg[i];
    if (residual) v = elu1(v + hin[i]);
    agg[i] = elu1(v);
}

// hg[g,:] += h[n,:] / 320
__global__ void pool_mean(const float* __restrict__ h,
                          const int* __restrict__ gid,
                          float* __restrict__ hg) {
    size_t i = (size_t)blockIdx.x * blockDim.x + threadIdx.x;  // over N*HD
    if (i >= (size_t)NN * HD) return;
    int n = (int)(i >> 8);
    int c = (int)(i & 255);
    atomicAdd(&hg[gid[n] * HD + c], h[i] * (1.0f / NODES_PER_G));
}

// out[g,c] = elu(hg[g,:]) @ Wc[:,c] + bc[c]
__global__ void classify(const float* __restrict__ hg,
                         const float* __restrict__ Wc,
                         const float* __restrict__ bc,
                         float* __restrict__ out) {
    int idx = blockIdx.x * blockDim.x + threadIdx.x;
    if (idx >= GG * CLS) return;
    int g = idx / CLS, c = idx % CLS;
    float acc = bc[c];
    for (int k = 0; k < HD; ++k) {
        acc += elu1(hg[g * HD + k]) * Wc[k * CLS + c];
    }
    out[idx] = acc;
}

// ---------------- launch ----------------
extern "C" void kernel_launch(void* const* d_in, const int* in_sizes, int n_in,
                              void* d_out, int out_size, void* d_ws, size_t ws_size,
                              hipStream_t stream) {
    (void)in_sizes; (void)n_in; (void)out_size; (void)ws_size;

    const float* x   = (const float*)d_in[0];
    const int*   src = (const int*)  d_in[1];
    const int*   dst = (const int*)  d_in[2];
    const int*   gid = (const int*)  d_in[3];
    const float* W0  = (const float*)d_in[4];
    const float* al0 = (const float*)d_in[5];
    const float* ar0 = (const float*)d_in[6];
    const float* W1  = (const float*)d_in[7];
    const float* al1 = (const float*)d_in[8];
    const float* ar1 = (const float*)d_in[9];
    const float* W2  = (const float*)d_in[10];
    const float* al2 = (const float*)d_in[11];
    const float* ar2 = (const float*)d_in[12];
    const float* Wc  = (const float*)d_in[13];
    const float* bc  = (const float*)d_in[14];
    float* out = (float*)d_out;

    // workspace layout (floats)
    float* ws = (float*)d_ws;
    float* B0   = ws;                        // N*HD  (z scratch)
    float* B1   = B0 + (size_t)NN * HD;      // N*HD  (agg/h ping)
    float* B2   = B1 + (size_t)NN * HD;      // N*HD  (agg/h pong)
    float* Ebuf = B2 + (size_t)NN * HD;      // E*H
    float* el   = Ebuf + (size_t)EE * HEADS; // N*H
    float* er   = el + (size_t)NN * HEADS;
    float* mb   = er + (size_t)NN * HEADS;
    float* sb   = mb + (size_t)NN * HEADS;
    float* hg   = sb + (size_t)NN * HEADS;   // G*HD

    const int NH  = NN * HEADS;              // 81920
    const int EH  = EE * HEADS;              // 1310720
    const int NHD_BLKS = (NN * HD) / 256;    // 20480

    auto run_layer = [&](const float* hin, int K, const float* W,
                         const float* al, const float* ar,
                         float* Z, float* AGG, int residual) {
        // z = hin @ W  (WMMA fp32, async-LDS staged B)
        gemm_wmma_f32<<<NN / 16, 128, 0, stream>>>(hin, W, Z, NN, K);
        // per-(node,head) attention logits
        attn_logits<<<(NH * 32) / 256, 256, 0, stream>>>(Z, al, ar, el, er);
        // init reductions
        fill_f32<<<(NH + 255) / 256, 256, 0, stream>>>(mb, -INFINITY, NH);
        hipMemsetAsync(sb, 0, (size_t)NH * sizeof(float), stream);
        hipMemsetAsync(AGG, 0, (size_t)NN * HD * sizeof(float), stream);
        // edge softmax
        edge_logits_max<<<(EH + 255) / 256, 256, 0, stream>>>(src, dst, el, er, Ebuf, mb);
        edge_exp_sum<<<(EH + 255) / 256, 256, 0, stream>>>(dst, mb, Ebuf, sb);
        // scatter messages
        edge_message<<<EE, 256, 0, stream>>>(src, dst, Z, Ebuf, sb, AGG);
        // activation (+ residual) in place -> AGG is the layer output
        act_residual<<<NHD_BLKS, 256, 0, stream>>>(hin, AGG, residual);
    };

    run_layer(x,  IN_DIM, W0, al0, ar0, B0, B1, 0);   // h1 = B1
    run_layer(B1, HD,     W1, al1, ar1, B0, B2, 1);   // h2 = B2
    run_layer(B2, HD,     W2, al2, ar2, B0, B1, 1);   // h3 = B1

    // readout
    hipMemsetAsync(hg, 0, (size_t)GG * HD * sizeof(float), stream);
    pool_mean<<<NHD_BLKS, 256, 0, stream>>>(B1, gid, hg);
    classify<<<(GG * CLS + 255) / 256, 256, 0, stream>>>(hg, Wc, bc, out);
}